// GNN_17575006175684
// MI455X (gfx1250) — compile-verified
//
#include <hip/hip_runtime.h>
#include <math.h>

#define NN 100000
#define NE 1600000
#define NG 1024

typedef __attribute__((ext_vector_type(2))) float v2f;
typedef __attribute__((ext_vector_type(8))) float v8f;

// ---------------- utility kernels ----------------
__global__ void zero_f32(float* __restrict__ p, int n) {
  int i = blockIdx.x * blockDim.x + threadIdx.x;
  if (i < n) p[i] = 0.0f;
}

__global__ void zero_u32(unsigned* __restrict__ p, int n) {
  int i = blockIdx.x * blockDim.x + threadIdx.x;
  if (i < n) p[i] = 0u;
}

__global__ void deg_count(const int* __restrict__ dst, unsigned* __restrict__ deg, int E) {
  int e = blockIdx.x * blockDim.x + threadIdx.x;
  if (e < E) atomicAdd(&deg[dst[e]], 1u);
}

__global__ void dinv_kernel(const unsigned* __restrict__ deg, float* __restrict__ dinv, int n) {
  int i = blockIdx.x * blockDim.x + threadIdx.x;
  if (i < n) dinv[i] = rsqrtf((float)deg[i] + 1.0f);  // +1 self loop, deg>=1 safe
}

// ---------------- WMMA GEMM: H[M,NOUT] = act(X)[M,K] @ W[K,NOUT] ----------------
// 4 waves per block, one wave32 per 16-row tile. f32 WMMA 16x16x4, K zero-padded.
// A layout: lanes 0-15 hold K pair {0,1}; lanes 16-31 hold K pair {2,3}.
// C layout: VGPR i = row (i + 8*(lane>=16)), col = lane&15.
// RELU fuses the previous layer's activation into the A-fragment load.
template <int K, int NOUT, bool RELU>
__global__ __launch_bounds__(128) void gemm_wmma(const float* __restrict__ X,
                                                 const float* __restrict__ W,
                                                 float* __restrict__ H,
                                                 int ntiles) {
  const int tile = blockIdx.x * 4 + (threadIdx.x >> 5);  // wave-uniform
  if (tile >= ntiles) return;                            // whole-wave exit, EXEC stays all-1s
  const int lane  = threadIdx.x & 31;
  const int m     = lane & 15;        // A row within tile / B,C column
  const int khalf = lane >> 4;        // 0 -> K {0,1}, 1 -> K {2,3}
  const int row0  = tile * 16;
  constexpr int KSTEPS = (K + 3) / 4;

  // preload A fragments for all K-steps (reused across N tiles)
  v2f a[KSTEPS];
  const float* xrow = X + (size_t)(row0 + m) * K;
#pragma unroll
  for (int ks = 0; ks < KSTEPS; ++ks) {
    const int k0 = ks * 4 + khalf * 2;
    float a0 = (k0     < K) ? xrow[k0]     : 0.0f;
    float a1 = (k0 + 1 < K) ? xrow[k0 + 1] : 0.0f;
    if (RELU) { a0 = fmaxf(a0, 0.0f); a1 = fmaxf(a1, 0.0f); }
    a[ks].x = a0;
    a[ks].y = a1;
  }

#pragma unroll
  for (int nt = 0; nt < NOUT / 16; ++nt) {
    v8f c = {};
#pragma unroll
    for (int ks = 0; ks < KSTEPS; ++ks) {
      const int k0 = ks * 4 + khalf * 2;
      v2f b;
      b.x = (k0     < K) ? W[(size_t)(k0)     * NOUT + nt * 16 + m] : 0.0f;
      b.y = (k0 + 1 < K) ? W[(size_t)(k0 + 1) * NOUT + nt * 16 + m] : 0.0f;
      c = __builtin_amdgcn_wmma_f32_16x16x4_f32(false, a[ks], false, b,
                                                (short)0, c, false, false);
    }
    const int col = nt * 16 + m;
#pragma unroll
    for (int i = 0; i < 8; ++i) {
      const int r = row0 + i + khalf * 8;
      H[(size_t)r * NOUT + col] = c[i];
    }
  }
}

// ---------------- GCN aggregation ----------------
// agg[i,f] = h[i,f]*dinv[i]^2 + bias[f]   (self-loop term + bias, full overwrite)
template <int F>
__global__ void self_init(const float* __restrict__ h, const float* __restrict__ dinv,
                          const float* __restrict__ bias, float* __restrict__ agg, int n) {
  int idx = blockIdx.x * blockDim.x + threadIdx.x;
  if (idx < n * F) {
    const int i = idx / F, f = idx & (F - 1);  // F = power of two -> shift/mask
    const float di = dinv[i];
    agg[idx] = h[idx] * di * di + bias[f];
  }
}

// one thread per (edge, feature): agg[dst,f] += h[src,f] * dinv[src]*dinv[dst]
// consecutive lanes share an edge -> gather/scatter hits 1-2 cachelines per wave
template <int F>
__global__ void edge_agg(const float* __restrict__ h, const float* __restrict__ dinv,
                         const int* __restrict__ src, const int* __restrict__ dst,
                         float* __restrict__ agg, int E) {
  int idx = blockIdx.x * blockDim.x + threadIdx.x;  // E*F <= 102.4M < 2^31
  if (idx < E * F) {
    const int e = idx / F, f = idx & (F - 1);       // shift/mask
    const int s = src[e], d = dst[e];
    const float norm = dinv[s] * dinv[d];
    atomicAdd(&agg[(size_t)d * F + f], h[(size_t)s * F + f] * norm);
  }
}

// ---------------- pooling + FC + log_softmax ----------------
__global__ void graph_count(const int* __restrict__ batch, float* __restrict__ counts, int n) {
  int i = blockIdx.x * blockDim.x + threadIdx.x;
  if (i < n) atomicAdd(&counts[batch[i]], 1.0f);
}

// fuses the final ReLU (act is stored pre-activation)
__global__ void graph_pool(const float* __restrict__ act, const int* __restrict__ batch,
                           float* __restrict__ pooled, int n) {
  int idx = blockIdx.x * blockDim.x + threadIdx.x;  // n*64
  if (idx < n * 64) {
    const int i = idx >> 6, f = idx & 63;
    atomicAdd(&pooled[(size_t)batch[i] * 64 + f], fmaxf(act[idx], 0.0f));
  }
}

__global__ void fc_logsoftmax(const float* __restrict__ pooled, const float* __restrict__ counts,
                              const float* __restrict__ Wfc, const float* __restrict__ bfc,
                              float* __restrict__ out, int G) {
  int g = blockIdx.x * blockDim.x + threadIdx.x;
  if (g < G) {
    const float inv = 1.0f / fmaxf(counts[g], 1.0f);
    float l0 = bfc[0], l1 = bfc[1];
    const float* pg = pooled + (size_t)g * 64;
#pragma unroll 8
    for (int f = 0; f < 64; ++f) {
      const float p = pg[f] * inv;
      l0 = fmaf(p, Wfc[f * 2 + 0], l0);
      l1 = fmaf(p, Wfc[f * 2 + 1], l1);
    }
    const float mx  = fmaxf(l0, l1);
    const float lse = mx + logf(__expf(l0 - mx) + __expf(l1 - mx));
    out[g * 2 + 0] = l0 - lse;
    out[g * 2 + 1] = l1 - lse;
  }
}

// ---------------- launch ----------------
static inline int cdiv(int a, int b) { return (a + b - 1) / b; }

extern "C" void kernel_launch(void* const* d_in, const int* in_sizes, int n_in,
                              void* d_out, int out_size, void* d_ws, size_t ws_size,
                              hipStream_t stream) {
  (void)in_sizes; (void)n_in; (void)out_size; (void)ws_size;

  const float* x     = (const float*)d_in[0];   // [NN,5]
  const int*   src   = (const int*)  d_in[1];   // [NE]
  const int*   dst   = (const int*)  d_in[2];   // [NE]
  const int*   batch = (const int*)  d_in[3];   // [NN]
  const float* W1    = (const float*)d_in[4];   // [5,16]
  const float* b1    = (const float*)d_in[5];
  const float* W2    = (const float*)d_in[6];   // [16,32]
  const float* b2    = (const float*)d_in[7];
  const float* W3    = (const float*)d_in[8];   // [32,64]
  const float* b3    = (const float*)d_in[9];
  const float* Wfc   = (const float*)d_in[10];  // [64,2]
  const float* bfc   = (const float*)d_in[11];
  float* out = (float*)d_out;                   // [NG,2]

  // workspace layout (floats)
  float*    ws     = (float*)d_ws;
  float*    dinv   = ws;                         // NN
  float*    bufH   = dinv  + NN;                 // NN*64 (GEMM output h)
  float*    bufAct = bufH  + (size_t)NN * 64;    // NN*64 (pre-activation agg)
  float*    pooled = bufAct + (size_t)NN * 64;   // NG*64
  float*    counts = pooled + (size_t)NG * 64;   // NG
  unsigned* deg    = (unsigned*)(counts + NG);   // NN u32

  const int B = 256;
  const int tiles = NN / 16;          // 6250 exactly
  const int gemmBlocks = cdiv(tiles, 4);

  // degree + dinv (shared by all layers)
  zero_u32<<<cdiv(NN, B), B, 0, stream>>>(deg, NN);
  deg_count<<<cdiv(NE, B), B, 0, stream>>>(dst, deg, NE);
  dinv_kernel<<<cdiv(NN, B), B, 0, stream>>>(deg, dinv, NN);

  // ---- layer 1: 5 -> 16 (input is raw x, no ReLU) ----
  gemm_wmma<5, 16, false><<<gemmBlocks, 128, 0, stream>>>(x, W1, bufH, tiles);
  self_init<16><<<cdiv(NN * 16, B), B, 0, stream>>>(bufH, dinv, b1, bufAct, NN);
  edge_agg<16><<<cdiv(NE * 16, B), B, 0, stream>>>(bufH, dinv, src, dst, bufAct, NE);

  // ---- layer 2: 16 -> 32 (ReLU fused into A-fragment load) ----
  gemm_wmma<16, 32, true><<<gemmBlocks, 128, 0, stream>>>(bufAct, W2, bufH, tiles);
  self_init<32><<<cdiv(NN * 32, B), B, 0, stream>>>(bufH, dinv, b2, bufAct, NN);
  edge_agg<32><<<cdiv(NE * 32, B), B, 0, stream>>>(bufH, dinv, src, dst, bufAct, NE);

  // ---- layer 3: 32 -> 64 ----
  gemm_wmma<32, 64, true><<<gemmBlocks, 128, 0, stream>>>(bufAct, W3, bufH, tiles);
  self_init<64><<<cdiv(NN * 64, B), B, 0, stream>>>(bufH, dinv, b3, bufAct, NN);
  edge_agg<64><<<cdiv(NE * 64, B), B, 0, stream>>>(bufH, dinv, src, dst, bufAct, NE);

  // ---- mean pool (ReLU fused) + FC + log_softmax ----
  zero_f32<<<cdiv(NG * 64, B), B, 0, stream>>>(pooled, NG * 64);
  zero_f32<<<cdiv(NG, B), B, 0, stream>>>(counts, NG);
  graph_count<<<cdiv(NN, B), B, 0, stream>>>(batch, counts, NN);
  graph_pool<<<cdiv(NN * 64, B), B, 0, stream>>>(bufAct, batch, pooled, NN);
  fc_logsoftmax<<<cdiv(NG, 64), 64, 0, stream>>>(pooled, counts, Wfc, bfc, out, NG);
}